// MyContrastiveLoss_24378234372803
// MI455X (gfx1250) — compile-verified
//
#include <hip/hip_runtime.h>
#include <hip/hip_bf16.h>
#include <stdint.h>

#define NUM_CLASSES 50
#define TEMP_W 0.1f
#define B_ROWS 8192
#define D_DIM 2048

typedef float v2f __attribute__((ext_vector_type(2)));
typedef float v8f __attribute__((ext_vector_type(8)));

// ---------------- threefry2x32 (key = (0,42), counter = (i,j)) ----------------
__device__ __forceinline__ uint32_t rotl32(uint32_t x, int r) {
  return (x << r) | (x >> (32 - r));
}

__device__ __forceinline__ uint32_t threefry2x32_x0(uint32_t k0, uint32_t k1,
                                                    uint32_t c0, uint32_t c1) {
  uint32_t ks2 = k0 ^ k1 ^ 0x1BD11BDAu;
  uint32_t x0 = c0 + k0, x1 = c1 + k1;
#define TF_R4(a, b, c, d)                         \
  x0 += x1; x1 = rotl32(x1, a); x1 ^= x0;         \
  x0 += x1; x1 = rotl32(x1, b); x1 ^= x0;         \
  x0 += x1; x1 = rotl32(x1, c); x1 ^= x0;         \
  x0 += x1; x1 = rotl32(x1, d); x1 ^= x0;
  TF_R4(13, 15, 26, 6)  x0 += k1;  x1 += ks2 + 1u;
  TF_R4(17, 29, 16, 24) x0 += ks2; x1 += k0 + 2u;
  TF_R4(13, 15, 26, 6)  x0 += k0;  x1 += k1 + 3u;
  TF_R4(17, 29, 16, 24) x0 += k1;  x1 += ks2 + 4u;
  TF_R4(13, 15, 26, 6)  x0 += ks2; x1 += k0 + 5u;
#undef TF_R4
  return x0;
}

// ---------------- kernel 1: one wave per row i, scan all j for argmax --------
__global__ void pair_select_kernel(const int* __restrict__ labels,
                                   int* __restrict__ pair) {
  __shared__ int lds_labels[B_ROWS];  // 32 KB of the WGP's 320 KB LDS
  for (int t = threadIdx.x; t < B_ROWS; t += blockDim.x)
    lds_labels[t] = labels[t];
  __syncthreads();

  const int wave = threadIdx.x >> 5;
  const int lane = threadIdx.x & 31;
  const int i = blockIdx.x * (blockDim.x >> 5) + wave;  // row handled by wave
  const int li = lds_labels[i];

  uint32_t best_u = 0u;
  int best_j = -1;
  // each lane scans j = lane, lane+32, ... (256 iterations)
  for (int s = 0; s < B_ROWS / 32; ++s) {
    const int j = s * 32 + lane;
    const int lj = lds_labels[j];
    if (lj == li && j != i && j != li) {
      // argmax of raw threefry bits == argmax of gumbel (monotone transform)
      const uint32_t u = threefry2x32_x0(0u, 42u, (uint32_t)i, (uint32_t)j);
      if (best_j < 0 || u > best_u || (u == best_u && j < best_j)) {
        best_u = u;
        best_j = j;
      }
    }
  }
  // wave-wide argmax merge (wave32)
  for (int off = 16; off > 0; off >>= 1) {
    const uint32_t ou = (uint32_t)__shfl_xor((int)best_u, off, 32);
    const int oj = __shfl_xor(best_j, off, 32);
    if (oj >= 0 &&
        (best_j < 0 || ou > best_u || (ou == best_u && oj < best_j))) {
      best_u = ou;
      best_j = oj;
    }
  }
  if (lane == 0) pair[i] = best_j;
}

// ---------------- kernel 2: ||pre_i - pre_j||, one wave per row --------------
// Streaming b128 loads (memory-bound: ~128 MB total, pre fits in 192MB L2),
// per-lane FMA accumulation, then V_WMMA_F32_16X16X4_F32 used as a 32->16
// cross-lane reduction (A selects K=0 and K=2: D[m,n] = p_n + p_{n+16}).
__global__ void pair_dist_kernel(const float* __restrict__ pre,
                                 const int* __restrict__ pair,
                                 float* __restrict__ dist) {
  const int wave = threadIdx.x >> 5;
  const int lane = threadIdx.x & 31;
  const int r = blockIdx.x * (blockDim.x >> 5) + wave;

  const int j = pair[r];  // wave-uniform -> EXEC stays all-ones for the WMMA
  float p = 0.0f;
  if (j >= 0) {
    const float4* __restrict__ xi = (const float4*)(pre + (size_t)r * D_DIM);
    const float4* __restrict__ xj = (const float4*)(pre + (size_t)j * D_DIM);
    float4 acc = make_float4(0.f, 0.f, 0.f, 0.f);
#pragma unroll 4
    for (int s = 0; s < D_DIM / 4 / 32; ++s) {  // 16 coalesced b128 steps
      const float4 a = xi[s * 32 + lane];
      const float4 b = xj[s * 32 + lane];
      const float dx = a.x - b.x, dy = a.y - b.y, dz = a.z - b.z, dw = a.w - b.w;
      acc.x = fmaf(dx, dx, acc.x);
      acc.y = fmaf(dy, dy, acc.y);
      acc.z = fmaf(dz, dz, acc.z);
      acc.w = fmaf(dw, dw, acc.w);
    }
    p = (acc.x + acc.y) + (acc.z + acc.w);
  }

  // WMMA reduction: A[m,0]=A[m,2]=1, A[m,1]=A[m,3]=0; B VGPR0 = partial p.
  // D[m,n] = B[0,n] + B[2,n] = p_n + p_{n+16} for every lane's d[0].
  v2f av;
  av[0] = 1.0f;
  av[1] = 0.0f;
  v2f bv;
  bv[0] = p;
  bv[1] = 0.0f;
  v8f c = {};
  c = __builtin_amdgcn_wmma_f32_16x16x4_f32(
      /*neg_a=*/false, av, /*neg_b=*/false, bv,
      /*c_mod=*/(short)0, c, /*reuse_a=*/false, /*reuse_b=*/false);
  float red = c[0];  // 16 partials left; finish with 4 xor-shuffles
  red += __shfl_xor(red, 1, 32);
  red += __shfl_xor(red, 2, 32);
  red += __shfl_xor(red, 4, 32);
  red += __shfl_xor(red, 8, 32);

  if (lane == 0) dist[r] = (j >= 0) ? sqrtf(red) : 0.0f;
}

// ---------------- kernel 3: cross-entropy, one wave per row ------------------
__global__ void ce_kernel(const float* __restrict__ post,
                          const int* __restrict__ labels,
                          float* __restrict__ ce) {
  const int wave = threadIdx.x >> 5;
  const int lane = threadIdx.x & 31;
  const int r = blockIdx.x * (blockDim.x >> 5) + wave;
  const float* __restrict__ row = post + (size_t)r * NUM_CLASSES;

  const float neg_inf = -__builtin_inff();
  const float x0 = (lane < NUM_CLASSES) ? row[lane] : neg_inf;
  const float x1 = (lane + 32 < NUM_CLASSES) ? row[lane + 32] : neg_inf;

  float m = fmaxf(x0, x1);
  for (int off = 16; off > 0; off >>= 1) m = fmaxf(m, __shfl_xor(m, off, 32));

  float s = 0.0f;
  if (lane < NUM_CLASSES) s += __expf(x0 - m);
  if (lane + 32 < NUM_CLASSES) s += __expf(x1 - m);
  for (int off = 16; off > 0; off >>= 1) s += __shfl_xor(s, off, 32);

  if (lane == 0) {
    const int lab = labels[r];
    ce[r] = (__logf(s) + m) - row[lab];
  }
}

// ---------------- kernel 4: deterministic final reduction --------------------
__global__ void final_reduce_kernel(const float* __restrict__ dist,
                                    const float* __restrict__ ce,
                                    float* __restrict__ out) {
  __shared__ float s_d[256];
  __shared__ float s_c[256];
  float sd = 0.0f, sc = 0.0f;
  for (int t = threadIdx.x; t < B_ROWS; t += 256) {
    sd += dist[t];
    sc += ce[t];
  }
  s_d[threadIdx.x] = sd;
  s_c[threadIdx.x] = sc;
  __syncthreads();
  for (int off = 128; off > 0; off >>= 1) {
    if (threadIdx.x < off) {
      s_d[threadIdx.x] += s_d[threadIdx.x + off];
      s_c[threadIdx.x] += s_c[threadIdx.x + off];
    }
    __syncthreads();
  }
  if (threadIdx.x == 0) {
    const float inv_b = 1.0f / (float)B_ROWS;
    out[0] = s_c[0] * inv_b + TEMP_W * (s_d[0] * inv_b);
  }
}

extern "C" void kernel_launch(void* const* d_in, const int* in_sizes, int n_in,
                              void* d_out, int out_size, void* d_ws,
                              size_t ws_size, hipStream_t stream) {
  const float* pre = (const float*)d_in[0];     // [8192, 2048] fp32
  const float* post = (const float*)d_in[1];    // [8192, 50] fp32
  const int* labels = (const int*)d_in[2];      // [8192] int32
  float* out = (float*)d_out;                   // scalar fp32

  int* pair = (int*)d_ws;                               // 32 KB
  float* dist = (float*)((char*)d_ws + B_ROWS * 4);     // 32 KB
  float* ce = dist + B_ROWS;                            // 32 KB

  // 8 waves per block, one wave per row
  pair_select_kernel<<<B_ROWS / 8, 256, 0, stream>>>(labels, pair);
  pair_dist_kernel<<<B_ROWS / 8, 256, 0, stream>>>(pre, pair, dist);
  ce_kernel<<<B_ROWS / 8, 256, 0, stream>>>(post, labels, ce);
  final_reduce_kernel<<<1, 256, 0, stream>>>(dist, ce, out);
}